// MaskLoss_1580547974629
// MI455X (gfx1250) — compile-verified
//
#include <hip/hip_runtime.h>
#include <math.h>

typedef __attribute__((ext_vector_type(2))) float v2f;
typedef __attribute__((ext_vector_type(8))) float v8f;

#define STEEP 100.0f
#define THRESH 0.55f
#define TILE_COLS 128
#define LDS_STRIDE 132   // 16B-aligned padding
#define COL_SLICES 8

__device__ __forceinline__ float fast_rcp(float x) {
    return __builtin_amdgcn_rcpf(x);
}

// ---------------- Kernel 1a: partial column max (row-sliced for occupancy) ----
__global__ __launch_bounds__(128) void colmax_partial(const float* __restrict__ tgt,
                                                      float* __restrict__ part,
                                                      int N, int W, int rowsPerSlice) {
    const int j  = blockIdx.x * 128 + threadIdx.x;
    const int s  = blockIdx.y;
    const int r0 = s * rowsPerSlice;
    const int r1 = min(N, r0 + rowsPerSlice);
    float m = -INFINITY;
    for (int i = r0; i < r1; ++i)
        m = fmaxf(m, tgt[(size_t)i * W + j]);   // coalesced across lanes
    part[(size_t)s * W + j] = m;
}

// ---------------- Kernel 1b: reduce slice maxima ------------------------------
__global__ __launch_bounds__(128) void colmax_reduce(const float* __restrict__ part,
                                                     float* __restrict__ values,
                                                     int W, int nSlices) {
    const int j = blockIdx.x * 128 + threadIdx.x;
    if (j >= W) return;
    float m = -INFINITY;
    for (int s = 0; s < nSlices; ++s)
        m = fmaxf(m, part[(size_t)s * W + j]);
    values[j] = m;
}

// ---------------- Kernel 2: per-row num / row_in / row_tg via WMMA row-sums ---
// Double-buffered LDS tiles; f32 WMMA (16x16x4) against all-ones B gives exact
// f32 row sums for 16 rows at a time per wave.
__global__ __launch_bounds__(256) void rowstats_wmma(const float* __restrict__ inp,
                                                     const float* __restrict__ tgt,
                                                     const float* __restrict__ values,
                                                     float* __restrict__ num,
                                                     float* __restrict__ rin,
                                                     float* __restrict__ rtg,
                                                     int N, int W) {
    __shared__ float s_ab[2][16][LDS_STRIDE];
    __shared__ float s_in[2][16][LDS_STRIDE];
    __shared__ float s_tg[2][16][LDS_STRIDE];
    __shared__ float s_val[16];
    __shared__ float s_part[3][8][16];

    const int tid  = threadIdx.x;
    const int lane = tid & 31;
    const int wv   = tid >> 5;         // 8 waves
    const int row0 = blockIdx.x * 16;

    if (tid < 16) s_val[tid] = values[row0 + tid];
    __syncthreads();

    // Per-thread tile-slot mapping: two float4 slots per thread
    int r_[2], cv_[2];
#pragma unroll
    for (int s = 0; s < 2; ++s) {
        const int linear = tid + s * 256;   // 0..511
        r_[s]  = linear >> 5;               // row 0..15
        cv_[s] = linear & 31;               // float4 col slot 0..31
    }

    // ---- prologue: stage chunk 0 into buffer 0 ----
#pragma unroll
    for (int s = 0; s < 2; ++s) {
        const int gcol = cv_[s] * 4;
        const size_t base = (size_t)(row0 + r_[s]) * W + gcol;
        const float4 vi = *(const float4*)(inp + base);
        const float4 vt = *(const float4*)(tgt + base);
        const float thr = THRESH * s_val[r_[s]];
        float4 ab;
        ab.x = fminf(vi.x, vt.x) * fast_rcp(1.0f + __expf(-STEEP * (vt.x - thr)));
        ab.y = fminf(vi.y, vt.y) * fast_rcp(1.0f + __expf(-STEEP * (vt.y - thr)));
        ab.z = fminf(vi.z, vt.z) * fast_rcp(1.0f + __expf(-STEEP * (vt.z - thr)));
        ab.w = fminf(vi.w, vt.w) * fast_rcp(1.0f + __expf(-STEEP * (vt.w - thr)));
        *(float4*)&s_ab[0][r_[s]][cv_[s] * 4] = ab;
        *(float4*)&s_in[0][r_[s]][cv_[s] * 4] = vi;
        *(float4*)&s_tg[0][r_[s]][cv_[s] * 4] = vt;
    }
    __syncthreads();

    v8f c_ab = {}; v8f c_in = {}; v8f c_tg = {};
    const v2f b_ones = {1.0f, 1.0f};

    // A-matrix 16x4 f32 layout: lanes 0-15 hold K=0,1; lanes 16-31 hold K=2,3
    const int rowA = lane & 15;
    const int koff = (lane < 16) ? 0 : 2;

    int cur = 0;
    for (int cc = 0; cc < W; cc += TILE_COLS) {
        const int  nxt     = cur ^ 1;
        const bool hasNext = (cc + TILE_COLS) < W;

        // ---- issue global loads for next chunk early (overlap with WMMA) ----
        float4 vi[2], vt[2];
        if (hasNext) {
#pragma unroll
            for (int s = 0; s < 2; ++s) {
                const int gcol = cc + TILE_COLS + cv_[s] * 4;
                const size_t base = (size_t)(row0 + r_[s]) * W + gcol;
                vi[s] = *(const float4*)(inp + base);
                vt[s] = *(const float4*)(tgt + base);
            }
            if (cc + 2 * TILE_COLS < W) {   // global_prefetch_b8 two chunks out
                const size_t pf = (size_t)(row0 + r_[0]) * W + cc + 2 * TILE_COLS + cv_[0] * 4;
                __builtin_prefetch(inp + pf, 0, 1);
                __builtin_prefetch(tgt + pf, 0, 1);
            }
        }

        // ---- WMMA consume current buffer: wave wv owns cols [wv*16,wv*16+16) --
#pragma unroll
        for (int kk = 0; kk < 4; ++kk) {
            const int k0 = wv * 16 + kk * 4 + koff;
            v2f a;
            a.x = s_ab[cur][rowA][k0]; a.y = s_ab[cur][rowA][k0 + 1];
            c_ab = __builtin_amdgcn_wmma_f32_16x16x4_f32(false, a, false, b_ones,
                                                         (short)0, c_ab, false, false);
            a.x = s_in[cur][rowA][k0]; a.y = s_in[cur][rowA][k0 + 1];
            c_in = __builtin_amdgcn_wmma_f32_16x16x4_f32(false, a, false, b_ones,
                                                         (short)0, c_in, false, false);
            a.x = s_tg[cur][rowA][k0]; a.y = s_tg[cur][rowA][k0 + 1];
            c_tg = __builtin_amdgcn_wmma_f32_16x16x4_f32(false, a, false, b_ones,
                                                         (short)0, c_tg, false, false);
        }

        // ---- compute + stage next chunk into the other buffer ----
        if (hasNext) {
#pragma unroll
            for (int s = 0; s < 2; ++s) {
                const float thr = THRESH * s_val[r_[s]];
                float4 ab;
                ab.x = fminf(vi[s].x, vt[s].x) * fast_rcp(1.0f + __expf(-STEEP * (vt[s].x - thr)));
                ab.y = fminf(vi[s].y, vt[s].y) * fast_rcp(1.0f + __expf(-STEEP * (vt[s].y - thr)));
                ab.z = fminf(vi[s].z, vt[s].z) * fast_rcp(1.0f + __expf(-STEEP * (vt[s].z - thr)));
                ab.w = fminf(vi[s].w, vt[s].w) * fast_rcp(1.0f + __expf(-STEEP * (vt[s].w - thr)));
                *(float4*)&s_ab[nxt][r_[s]][cv_[s] * 4] = ab;
                *(float4*)&s_in[nxt][r_[s]][cv_[s] * 4] = vi[s];
                *(float4*)&s_tg[nxt][r_[s]][cv_[s] * 4] = vt[s];
            }
        }
        __syncthreads();   // single barrier per chunk
        cur = nxt;
    }

    // C layout: VGPR r, lanes 0-15 -> M=r; lanes 16-31 -> M=r+8. B=ones => all N equal.
    if (lane == 0) {
#pragma unroll
        for (int r = 0; r < 8; ++r) {
            s_part[0][wv][r] = c_ab[r];
            s_part[1][wv][r] = c_in[r];
            s_part[2][wv][r] = c_tg[r];
        }
    } else if (lane == 16) {
#pragma unroll
        for (int r = 0; r < 8; ++r) {
            s_part[0][wv][8 + r] = c_ab[r];
            s_part[1][wv][8 + r] = c_in[r];
            s_part[2][wv][8 + r] = c_tg[r];
        }
    }
    __syncthreads();

    if (tid < 16) {   // deterministic fixed-order cross-wave combine
        float a0 = 0.f, a1 = 0.f, a2 = 0.f;
#pragma unroll
        for (int w = 0; w < 8; ++w) {
            a0 += s_part[0][w][tid];
            a1 += s_part[1][w][tid];
            a2 += s_part[2][w][tid];
        }
        num[row0 + tid] = 2.0f * a0;
        rin[row0 + tid] = a1;
        rtg[row0 + tid] = a2;
    }
}

// ---------------- Kernel 3: pairwise sum over (i,k), block partials -----------
__global__ __launch_bounds__(256) void pairsum_kernel(const float* __restrict__ num,
                                                      const float* __restrict__ rin,
                                                      const float* __restrict__ rtg,
                                                      float* __restrict__ partials,
                                                      int N) {
    __shared__ float sred[256];
    const int tid = threadIdx.x;
    const int i0  = blockIdx.x * 16;

    float nu[16], ri[16];
#pragma unroll
    for (int ii = 0; ii < 16; ++ii) {
        nu[ii] = num[i0 + ii];
        ri[ii] = rin[i0 + ii];
    }

    float acc = 0.0f;
    for (int k0 = 0; k0 < N; k0 += 256) {
        const float t = rtg[k0 + tid];      // one load reused 16x from register
#pragma unroll
        for (int ii = 0; ii < 16; ++ii) {
            const float v = nu[ii] * fast_rcp(ri[ii] + t);
            acc += (v == v) ? v : 0.0f;     // NaN -> 0
        }
    }
    sred[tid] = acc;
    __syncthreads();
    for (int off = 128; off > 0; off >>= 1) {
        if (tid < off) sred[tid] += sred[tid + off];
        __syncthreads();
    }
    if (tid == 0) partials[blockIdx.x] = sred[0];
}

// ---------------- Kernel 4: final reduce + mean -------------------------------
__global__ __launch_bounds__(256) void finalize_kernel(const float* __restrict__ partials,
                                                       int nPart, float* __restrict__ out,
                                                       float invN2) {
    __shared__ float sred[256];
    const int tid = threadIdx.x;
    float a = 0.0f;
    for (int i = tid; i < nPart; i += 256) a += partials[i];
    sred[tid] = a;
    __syncthreads();
    for (int off = 128; off > 0; off >>= 1) {
        if (tid < off) sred[tid] += sred[tid + off];
        __syncthreads();
    }
    if (tid == 0) out[0] = sred[0] * invN2;
}

extern "C" void kernel_launch(void* const* d_in, const int* in_sizes, int n_in,
                              void* d_out, int out_size, void* d_ws, size_t ws_size,
                              hipStream_t stream) {
    const float* inp = (const float*)d_in[0];
    const float* tgt = (const float*)d_in[1];
    float* out = (float*)d_out;

    const int total = in_sizes[0];
    const int N = (int)(sqrt((double)total) + 0.5);   // 4096
    const int W = N;

    float* values   = (float*)d_ws;            // W
    float* num      = values + W;              // N
    float* rin      = num + N;                 // N
    float* rtg      = rin + N;                 // N
    float* partials = rtg + N;                 // N/16
    float* colpart  = partials + (N / 16);     // COL_SLICES * W

    const int rowsPerSlice = (N + COL_SLICES - 1) / COL_SLICES;
    dim3 gridCM(W / 128, COL_SLICES);
    colmax_partial<<<gridCM, 128, 0, stream>>>(tgt, colpart, N, W, rowsPerSlice);
    colmax_reduce<<<(W + 127) / 128, 128, 0, stream>>>(colpart, values, W, COL_SLICES);

    rowstats_wmma<<<N / 16, 256, 0, stream>>>(inp, tgt, values, num, rin, rtg, N, W);

    const int nPair = N / 16;
    pairsum_kernel<<<nPair, 256, 0, stream>>>(num, rin, rtg, partials, N);
    finalize_kernel<<<1, 256, 0, stream>>>(partials, nPair, out,
                                           1.0f / ((float)N * (float)N));
}